// POPMIProjection_70042326663932
// MI455X (gfx1250) — compile-verified
//
#include <hip/hip_runtime.h>
#include <stdint.h>

typedef unsigned int v4u __attribute__((ext_vector_type(4)));
typedef int          v8i __attribute__((ext_vector_type(8)));
typedef int          v4i __attribute__((ext_vector_type(4)));

#define NN 224
#define PLANE_ELEMS (NN * NN)      // 50176
#define TS 16
#define LSTR 18                    // 16 data dwords + 2 pad dwords inserted by TDM
#define TILE_DW (TS * LSTR)        // 288
#define TILE_BYTES (TILE_DW * 4)   // 1152

#if defined(__gfx1250__)
// TDM: DMA one 16x16 f32 tile (row-major, row stride 224 floats) from global
// into LDS at byte offset lds_off, inserting 2 pad dwords after every 64B row
// (-> LDS row stride = 18 dwords, conflict-free transposed reads).
__device__ __forceinline__ void tdm_load_tile(const float* gsrc, unsigned lds_off) {
  uint64_t ga = (uint64_t)(uintptr_t)gsrc;
  // D# group 0: count=1 | lds_addr | global_addr[56:0] | type=2 (bits 127:126)
  v4u g0 = {1u, lds_off, (unsigned)ga,
            ((unsigned)((ga >> 32) & 0x01FFFFFFull)) | 0x80000000u};
  // D# group 1:
  //  d0: wg_mask=0, data_size=2(4B), pad_enable=1, pad_interval=3(16dw), pad_amount=1(2dw)
  //  tensor_dim0=224, tensor_dim1=224, tile_dim0=16, tile_dim1=16, tile_dim2=0
  //  tensor_dim0_stride=224, tensor_dim1_stride=50176(=0xC400)
  v8i g1 = {(int)0x02D20000u,
            (int)(224u << 16),   // abar_addr=0 | tensor_dim0.lo16=224
            (int)(224u << 16),   // tensor_dim0.hi16=0 | tensor_dim1.lo16=224
            (int)(16u  << 16),   // tensor_dim1.hi16=0 | tile_dim0=16
            16,                  // tile_dim1=16 | tile_dim2=0
            224,                 // tensor_dim0_stride[31:0]
            (int)0xC4000000u,    // stride0[47:32]=0 | tensor_dim1_stride.lo16=0xC400
            0};                  // tensor_dim1_stride[47:16]
  // Unused upper dims: set lengths to 1 so index 0 stays in bounds, tiles 0.
  v4i g2 = {1, 1, 0, 0};               // tensor_dim2=1, tensor_dim3=1
  v4i g3 = {0, (int)(1u << 16), 0, 0}; // tensor_dim4=1
  v8i g4 = {0, 0, 0, 0, 0, 0, 0, 0};   // extra group in 6-arg builtin: zero-filled
  __builtin_amdgcn_tensor_load_to_lds(g0, g1, g2, g3, g4, 0);
}
#endif

// D4 symmetrization + residual mix, orbit-at-a-time.
// sigma(i,j)=(j,223-i) [rot90 CCW source map], tau(i,j)=(i,223-j).
// Base tile lives in the 112x112 quadrant; its 4 rotation images tile the plane.
__global__ __launch_bounds__(256) void d4_mix_kernel(const float* __restrict__ x,
                                                     const float* __restrict__ theta,
                                                     float* __restrict__ out) {
  __shared__ float lds[8 * TILE_DW];
  __shared__ float cf[9];

  const int tid   = threadIdx.x;
  const int bid   = blockIdx.x;
  const int plane = bid / 49;
  const int tl    = bid - plane * 49;
  const int ti    = tl / 7;
  const int tj    = tl - ti * 7;
  const int i0 = ti * 16, j0 = tj * 16;
  const int ra = 208 - i0, rb = 208 - j0;   // 208 = 223 - 15

  const size_t pbase = (size_t)plane * PLANE_ELEMS;
  const float* xp = x + pbase;

#if defined(__gfx1250__)
  {
    const int t = tid >> 5;            // one wave per source tile (8 waves)
    int r0, c0;
    switch (t) {                        // tile origins of sigma^r(p), sigma^r(tau(p))
      case 0:  r0 = i0; c0 = j0; break; // v0: identity
      case 1:  r0 = j0; c0 = ra; break; // v1: sigma
      case 2:  r0 = ra; c0 = rb; break; // v2: sigma^2
      case 3:  r0 = rb; c0 = i0; break; // v3: sigma^3
      case 4:  r0 = i0; c0 = rb; break; // u0: tau
      case 5:  r0 = rb; c0 = ra; break; // u1: sigma*tau
      case 6:  r0 = ra; c0 = j0; break; // u2: sigma^2*tau
      default: r0 = j0; c0 = i0; break; // u3: sigma^3*tau
    }
    const unsigned lbase = (unsigned)(uintptr_t)(&lds[0]);
    tdm_load_tile(xp + (size_t)r0 * NN + c0, lbase + (unsigned)t * TILE_BYTES);
  }
#else
  {
    const int trow[8] = {i0, j0, ra, rb, i0, rb, ra, j0};
    const int tcol[8] = {j0, ra, rb, i0, rb, ra, j0, i0};
    const int r = (tid >> 4) & 15, c = tid & 15;
    for (int t = 0; t < 8; ++t)
      lds[t * TILE_DW + r * LSTR + c] = xp[(size_t)(trow[t] + r) * NN + (tcol[t] + c)];
  }
#endif

  // softmax(theta) + scale, overlapped with the tensor DMA
  if (tid == 0) {
    float th[8];
    float m = -3.402823466e38f;
    for (int i = 0; i < 8; ++i) { th[i] = theta[i]; m = fmaxf(m, th[i]); }
    float e[8], s = 0.f;
    for (int i = 0; i < 8; ++i) { e[i] = expf(th[i] - m); s += e[i]; }
    float wsum = 0.f;
    for (int i = 0; i < 8; ++i) { float w = e[i] / s; cf[i] = w; wsum += w; }
    cf[8] = 0.5f / fmaxf(wsum, 1e-12f);   // BETA / max(sum(w), 1e-12)
  }

#if defined(__gfx1250__)
  __builtin_amdgcn_s_wait_tensorcnt(0);
#endif
  __syncthreads();

  const int tx = tid & 15;
  const int ty = (tid >> 4) & 15;
  const int sx = 15 - tx, sy = 15 - ty;

#define LT(t, r, c) lds[(t) * TILE_DW + (r) * LSTR + (c)]
  // v_r = x[sigma^r p], u_s = x[sigma^s tau p], in tile-local (transposed) coords
  const float v0 = LT(0, ty, tx);
  const float v1 = LT(1, tx, sy);
  const float v2 = LT(2, sy, sx);
  const float v3 = LT(3, sx, ty);
  const float u0 = LT(4, ty, sx);
  const float u1 = LT(5, sx, sy);
  const float u2 = LT(6, sy, tx);
  const float u3 = LT(7, tx, ty);

  const float w0 = cf[0], w1 = cf[1], w2 = cf[2], w3 = cf[3];
  const float w4 = cf[4], w5 = cf[5], w6 = cf[6], w7 = cf[7];
  const float bs = cf[8];

  // out at sigma^r p = 0.5*v_r + bs * sum_m w_m * x[g_m(sigma^r p)],
  // using tau sigma^r = sigma^{-r} tau; terms kept in the reference's element order.
  const float a0 = w0*v0 + w1*v1 + w2*v2 + w3*v3 + w4*u0 + w5*u3 + w6*u2 + w7*u1;
  const float a1 = w0*v1 + w1*v2 + w2*v3 + w3*v0 + w4*u3 + w5*u2 + w6*u1 + w7*u0;
  const float a2 = w0*v2 + w1*v3 + w2*v0 + w3*v1 + w4*u2 + w5*u1 + w6*u0 + w7*u3;
  const float a3 = w0*v3 + w1*v0 + w2*v1 + w3*v2 + w4*u1 + w5*u0 + w6*u3 + w7*u2;

  const float o0 = 0.5f * v0 + bs * a0;
  const float o1 = 0.5f * v1 + bs * a1;
  const float o2 = 0.5f * v2 + bs * a2;
  const float o3 = 0.5f * v3 + bs * a3;

  __syncthreads();                 // all LDS reads done before re-staging
  LT(0, ty, tx) = o0;              // stage transposed so stores are coalesced
  LT(1, tx, sy) = o1;
  LT(2, sy, sx) = o2;
  LT(3, sx, ty) = o3;
  __syncthreads();

  float* op = out + pbase;
  op[(size_t)(i0 + ty) * NN + (j0 + tx)] = LT(0, ty, tx);
  op[(size_t)(j0 + ty) * NN + (ra + tx)] = LT(1, ty, tx);
  op[(size_t)(ra + ty) * NN + (rb + tx)] = LT(2, ty, tx);
  op[(size_t)(rb + ty) * NN + (i0 + tx)] = LT(3, ty, tx);
#undef LT
}

extern "C" void kernel_launch(void* const* d_in, const int* in_sizes, int n_in,
                              void* d_out, int out_size, void* d_ws, size_t ws_size,
                              hipStream_t stream) {
  (void)n_in; (void)out_size; (void)d_ws; (void)ws_size;
  const float* x     = (const float*)d_in[0];
  const float* theta = (const float*)d_in[1];
  float* out = (float*)d_out;
  const int planes = in_sizes[0] / PLANE_ELEMS;            // 16*192 = 3072
  dim3 grid((unsigned)(planes * 49));                      // 49 quadrant tiles/plane
  d4_mix_kernel<<<grid, 256, 0, stream>>>(x, theta, out);
}